// TopoEdgeGNN_31344671326803
// MI455X (gfx1250) — compile-verified
//
#include <hip/hip_runtime.h>

typedef __attribute__((ext_vector_type(2))) float v2f;
typedef __attribute__((ext_vector_type(8))) float v8f;

#define N_NODES 100000
#define N_EDGES 1600000
#define DIM     128
#define EDIM    16
#define LAYERS  5
#define LN_EPS  1e-5f

// ---------------------------------------------------------------------------
// LayerNorm over feature dim (D=128). One wave32 per node, 4 floats per lane.
// Safe in-place: each wave reads its full row before any store.
// ---------------------------------------------------------------------------
__global__ __launch_bounds__(256) void ln_kernel(const float* __restrict__ x,
                                                 float* __restrict__ y,
                                                 const float* __restrict__ w,
                                                 const float* __restrict__ b) {
    int gwave = (blockIdx.x * blockDim.x + threadIdx.x) >> 5;
    int lane  = threadIdx.x & 31;
    if (gwave >= N_NODES) return;

    const float4* row = (const float4*)(x + (size_t)gwave * DIM);
    float4 v = row[lane];

    float sum = v.x + v.y + v.z + v.w;
#pragma unroll
    for (int off = 16; off > 0; off >>= 1) sum += __shfl_xor(sum, off);
    float mu = sum * (1.0f / DIM);

    float dx = v.x - mu, dy = v.y - mu, dz = v.z - mu, dw = v.w - mu;
    float ss = dx * dx + dy * dy + dz * dz + dw * dw;
#pragma unroll
    for (int off = 16; off > 0; off >>= 1) ss += __shfl_xor(ss, off);
    float inv = rsqrtf(ss * (1.0f / DIM) + LN_EPS);

    float4 wv = ((const float4*)w)[lane];
    float4 bv = ((const float4*)b)[lane];
    float4 o;
    o.x = dx * inv * wv.x + bv.x;
    o.y = dy * inv * wv.y + bv.y;
    o.z = dz * inv * wv.z + bv.z;
    o.w = dw * inv * wv.w + bv.w;
    ((float4*)(y + (size_t)gwave * DIM))[lane] = o;
}

// ---------------------------------------------------------------------------
// P = s @ W_a + eb1 ; Q = s @ W_b.  (N x 128) @ (128 x 16) -> (N x 16)
// One wave per 16-node tile; K=128 in 32 chunks of K=4 via
// V_WMMA_F32_16X16X4_F32. A (the s tile) is shared between both accumulators.
// ---------------------------------------------------------------------------
__global__ __launch_bounds__(256) void pq_kernel(const float* __restrict__ s,
                                                 const float* __restrict__ Wa,  // 128x16
                                                 const float* __restrict__ Wb,  // 128x16
                                                 const float* __restrict__ eb1, // 16
                                                 float* __restrict__ P,
                                                 float* __restrict__ Q) {
    int gwave = (blockIdx.x * blockDim.x + threadIdx.x) >> 5;
    int lane  = threadIdx.x & 31;
    if (gwave >= N_NODES / 16) return;

    int m    = lane & 15;     // A row index (M) and B/D column index (N)
    int hi   = lane >> 4;     // half-wave selector
    int base = gwave * 16;

    const float* srow = s + (size_t)(base + m) * DIM;

    v8f accP = {};
    v8f accQ = {};
#pragma unroll
    for (int k0 = 0; k0 < DIM; k0 += 4) {
        // A 16x4 f32 layout: v0 = K=k0 | K=k0+2, v1 = K=k0+1 | K=k0+3
        v2f a;
        a.x = srow[k0 + 2 * hi];
        a.y = srow[k0 + 2 * hi + 1];
        // B 4x16 f32 layout: v0 = row k0 | row k0+2, v1 = row k0+1 | row k0+3
        v2f bp, bq;
        bp.x = Wa[(k0 + 2 * hi) * 16 + m];
        bp.y = Wa[(k0 + 2 * hi + 1) * 16 + m];
        bq.x = Wb[(k0 + 2 * hi) * 16 + m];
        bq.y = Wb[(k0 + 2 * hi + 1) * 16 + m];
        accP = __builtin_amdgcn_wmma_f32_16x16x4_f32(false, a, false, bp,
                                                     (short)0, accP, false, false);
        accQ = __builtin_amdgcn_wmma_f32_16x16x4_f32(false, a, false, bq,
                                                     (short)0, accQ, false, false);
    }

    float bias = eb1[m];
#pragma unroll
    for (int j = 0; j < 8; ++j) {
        int r = base + j + 8 * hi;                 // D layout: VGPR j -> row j / j+8
        P[(size_t)r * 16 + m] = accP[j] + bias;
        Q[(size_t)r * 16 + m] = accQ[j];
    }
}

// ---------------------------------------------------------------------------
// Per-edge MLP: ea' = silu(P[src] + Q[dst] + ea@We) @ W2 + eb2.
// One wave per 16-edge tile; two 16x16 fp32 matmuls as 4+4 WMMA k-chunks.
// Gathered P/Q rows seed the first accumulator; SiLU'd hidden goes through a
// padded LDS transpose (intra-wave, DS in-order) to become the second A.
// ---------------------------------------------------------------------------
__global__ __launch_bounds__(256) void edge_kernel(const float* __restrict__ ea_in,
                                                   float* __restrict__ ea_out,
                                                   const long long* __restrict__ src,
                                                   const long long* __restrict__ dst,
                                                   const float* __restrict__ P,
                                                   const float* __restrict__ Q,
                                                   const float* __restrict__ We,  // 16x16
                                                   const float* __restrict__ W2,  // 16x16
                                                   const float* __restrict__ eb2) {
    __shared__ float tr[8][16][17];  // stride-17 padding: conflict-free transpose

    int gwave = (blockIdx.x * blockDim.x + threadIdx.x) >> 5;
    int wv    = (threadIdx.x >> 5) & 7;
    int lane  = threadIdx.x & 31;
    if (gwave >= N_EDGES / 16) return;

    int m     = lane & 15;
    int hi    = lane >> 4;
    int ebase = gwave * 16;

    // Seed C with gathered P[src] + Q[dst] (rows are 64B, coalesced over lanes 0-15)
    v8f c;
#pragma unroll
    for (int j = 0; j < 8; ++j) {
        int e  = ebase + j + 8 * hi;
        int si = (int)src[e];
        int di = (int)dst[e];
        c[j] = P[(size_t)si * 16 + m] + Q[(size_t)di * 16 + m];
    }

    const float* earow = ea_in + (size_t)(ebase + m) * EDIM;
#pragma unroll
    for (int k0 = 0; k0 < 16; k0 += 4) {
        v2f a, b;
        a.x = earow[k0 + 2 * hi];
        a.y = earow[k0 + 2 * hi + 1];
        b.x = We[(k0 + 2 * hi) * 16 + m];
        b.y = We[(k0 + 2 * hi + 1) * 16 + m];
        c = __builtin_amdgcn_wmma_f32_16x16x4_f32(false, a, false, b,
                                                  (short)0, c, false, false);
    }

    // SiLU + write D-layout into LDS (row = j + 8*hi, col = m)
#pragma unroll
    for (int j = 0; j < 8; ++j) {
        float x = c[j];
        tr[wv][j + 8 * hi][m] = x * (1.0f / (1.0f + __expf(-x)));
    }

    // Second matmul: A = hidden (read back in A-layout), C seeded with eb2
    float b2 = eb2[m];
    v8f c2;
#pragma unroll
    for (int j = 0; j < 8; ++j) c2[j] = b2;

#pragma unroll
    for (int k0 = 0; k0 < 16; k0 += 4) {
        v2f a, b;
        a.x = tr[wv][m][k0 + 2 * hi];
        a.y = tr[wv][m][k0 + 2 * hi + 1];
        b.x = W2[(k0 + 2 * hi) * 16 + m];
        b.y = W2[(k0 + 2 * hi + 1) * 16 + m];
        c2 = __builtin_amdgcn_wmma_f32_16x16x4_f32(false, a, false, b,
                                                   (short)0, c2, false, false);
    }

#pragma unroll
    for (int j = 0; j < 8; ++j) {
        ea_out[(size_t)(ebase + j + 8 * hi) * 16 + m] = c2[j];
    }
}

// ---------------------------------------------------------------------------
extern "C" void kernel_launch(void* const* d_in, const int* in_sizes, int n_in,
                              void* d_out, int out_size, void* d_ws, size_t ws_size,
                              hipStream_t stream) {
    const float*     s_in  = (const float*)d_in[0];
    const long long* eidx  = (const long long*)d_in[1];  // int64 [2, E]
    const float*     ea_in = (const float*)d_in[2];
    // d_in[3] = batch (dead), d_in[10..13] = nW1/nb1/nW2/nb2 (dead: out discarded)
    const float* ln_w = (const float*)d_in[4];
    const float* ln_b = (const float*)d_in[5];
    const float* eW1  = (const float*)d_in[6];  // [L, 272, 16]
    const float* eb1  = (const float*)d_in[7];  // [L, 16]
    const float* eW2  = (const float*)d_in[8];  // [L, 16, 16]
    const float* eb2  = (const float*)d_in[9];  // [L, 16]

    float* out_s  = (float*)d_out;
    float* out_ea = out_s + (size_t)N_NODES * DIM;

    float* P = (float*)d_ws;                    // N x 16
    float* Q = P + (size_t)N_NODES * EDIM;      // N x 16

    const long long* srcI = eidx;
    const long long* dstI = eidx + N_EDGES;

    const int LN_BLOCKS   = (N_NODES * 32) / 256;          // 12500
    const int PQ_BLOCKS   = (N_NODES / 16 + 7) / 8;        // 782
    const int EDGE_BLOCKS = (N_EDGES / 16) / 8;            // 12500

    for (int i = 0; i < LAYERS; ++i) {
        const float* s_src  = (i == 0) ? s_in : out_s;
        const float* ea_src = (i == 0) ? ea_in : out_ea;   // elementwise per edge: in-place OK
        const float* W1     = eW1 + (size_t)i * 272 * 16;

        ln_kernel<<<LN_BLOCKS, 256, 0, stream>>>(s_src, out_s,
                                                 ln_w + (size_t)i * DIM,
                                                 ln_b + (size_t)i * DIM);
        pq_kernel<<<PQ_BLOCKS, 256, 0, stream>>>(out_s,
                                                 W1,                 // rows 0..127   (W_a)
                                                 W1 + 128 * 16,      // rows 128..255 (W_b)
                                                 eb1 + (size_t)i * 16,
                                                 P, Q);
        edge_kernel<<<EDGE_BLOCKS, 256, 0, stream>>>(ea_src, out_ea, srcI, dstI, P, Q,
                                                     W1 + 256 * 16,          // rows 256..271 (W_e)
                                                     eW2 + (size_t)i * 256,
                                                     eb2 + (size_t)i * 16);
    }
}